// GraphSeg_66425964200640
// MI455X (gfx1250) — compile-verified
//
#include <hip/hip_runtime.h>

// ---------------------------------------------------------------------------
// dense_mincut_pool for gfx1250 (CDNA5, wave32, WMMA f32 16x16x4)
// B=8, V=10000, D=128, K=64, E=320000
// ---------------------------------------------------------------------------

#define B_  8
#define V_  10000
#define D_  128
#define K_  64
#define E_  320000
#define VCHUNK 2000   // V_ / 5
#define ECHUNK 4000   // E_ / 80

typedef __attribute__((ext_vector_type(2))) float v2f;
typedef __attribute__((ext_vector_type(8))) float v8f;

// D(16x16,f32) = A(16x4,f32) x B(4x16,f32) + C
// A: lane L holds m=L%16, k = 2*(L/16)+{0,1} in the two VGPRs
// B: lane L holds n=L%16, k = 2*(L/16)+{0,1}
// C/D: vgpr r, lane L -> (m = r + 8*(L/16), n = L%16)
__device__ __forceinline__ v8f wmma4(v2f a, v2f b, v8f c) {
    return __builtin_amdgcn_wmma_f32_16x16x4_f32(
        /*neg_a=*/false, a, /*neg_b=*/false, b,
        /*c_mod=*/(short)0, c, /*reuse_a=*/false, /*reuse_b=*/false);
}

// ---------------------------------------------------------------------------
// K0: zero accumulators (ws prefix) and the x_pool output region
// ---------------------------------------------------------------------------
#define WS_FLOATS (V_ + (size_t)B_*V_ + 2*(size_t)B_*K_*K_ + 2*B_)

__global__ void k_init(float* __restrict__ ws, float* __restrict__ out_xp) {
    size_t i = (size_t)blockIdx.x * blockDim.x + threadIdx.x;
    if (i < WS_FLOATS) ws[i] = 0.0f;
    if (i < (size_t)B_ * K_ * D_) out_xp[i] = 0.0f;
}

// ---------------------------------------------------------------------------
// K1: d_flat[v] = segment_sum(edge_weight, row)   (batch independent)
// ---------------------------------------------------------------------------
__global__ void k_dflat(const int* __restrict__ ei, const float* __restrict__ ew,
                        float* __restrict__ dflat) {
    int t = blockIdx.x * blockDim.x + threadIdx.x;
    if (t < E_) unsafeAtomicAdd(&dflat[ei[t]], ew[t]);   // ei[0..E) = row
}

// ---------------------------------------------------------------------------
// K2: s = softmax(x @ W + bias), plus rowSumSq = sum_k s^2 (for mincut_den)
// One wave handles a 16(v) x 64(k) tile. 32 K-steps, 4 N-tiles.
// ---------------------------------------------------------------------------
__global__ __launch_bounds__(32)
void k_score_softmax(const float* __restrict__ x, const float* __restrict__ W,
                     const float* __restrict__ bias, float* __restrict__ s_out,
                     float* __restrict__ rss) {
    const int lane = threadIdx.x & 31;
    const int half = lane >> 4;
    const int l    = lane & 15;
    const int tile = blockIdx.x;                 // 0 .. B*(V/16)-1
    const int b    = tile / (V_ / 16);
    const int v0   = (tile - b * (V_ / 16)) * 16;

    const float* xA = x + ((size_t)b * V_ + v0 + l) * D_ + 2 * half;
    v8f acc0 = {}, acc1 = {}, acc2 = {}, acc3 = {};
    #pragma unroll 4
    for (int k0 = 0; k0 < D_; k0 += 4) {
        v2f a;  a.x = xA[k0];  a.y = xA[k0 + 1];
        const float* Wr = W + (size_t)(k0 + 2 * half) * K_ + l;
        v2f b0; b0.x = Wr[0];  b0.y = Wr[K_];
        v2f b1; b1.x = Wr[16]; b1.y = Wr[K_ + 16];
        v2f b2; b2.x = Wr[32]; b2.y = Wr[K_ + 32];
        v2f b3; b3.x = Wr[48]; b3.y = Wr[K_ + 48];
        acc0 = wmma4(a, b0, acc0);
        acc1 = wmma4(a, b1, acc1);
        acc2 = wmma4(a, b2, acc2);
        acc3 = wmma4(a, b3, acc3);
    }

    const float bb0 = bias[l], bb1 = bias[16 + l], bb2 = bias[32 + l], bb3 = bias[48 + l];
    #pragma unroll
    for (int r = 0; r < 8; ++r) {
        float e0 = acc0[r] + bb0, e1 = acc1[r] + bb1;
        float e2 = acc2[r] + bb2, e3 = acc3[r] + bb3;
        float mx = fmaxf(fmaxf(e0, e1), fmaxf(e2, e3));
        #pragma unroll
        for (int off = 8; off; off >>= 1) mx = fmaxf(mx, __shfl_xor(mx, off, 16));
        e0 = __expf(e0 - mx); e1 = __expf(e1 - mx);
        e2 = __expf(e2 - mx); e3 = __expf(e3 - mx);
        float sm = e0 + e1 + e2 + e3;
        #pragma unroll
        for (int off = 8; off; off >>= 1) sm += __shfl_xor(sm, off, 16);
        float inv = 1.0f / sm;
        e0 *= inv; e1 *= inv; e2 *= inv; e3 *= inv;
        float q = e0 * e0 + e1 * e1 + e2 * e2 + e3 * e3;
        #pragma unroll
        for (int off = 8; off; off >>= 1) q += __shfl_xor(q, off, 16);

        const int m = r + 8 * half;                 // row within tile
        float* dst = s_out + ((size_t)b * V_ + v0 + m) * K_ + l;
        dst[0] = e0; dst[16] = e1; dst[32] = e2; dst[48] = e3;
        if (l == 0) rss[(size_t)b * V_ + v0 + m] = q;
    }
}

// ---------------------------------------------------------------------------
// K3: x_pool += s^T x  over a V-chunk.  8 waves/block: wave w owns D-column
// tile w (n0 = 16w), 4 cluster m-tiles each. A gather is lane-contiguous
// because s is [v][k] row-major and m maps across lanes.
// ---------------------------------------------------------------------------
__global__ __launch_bounds__(256)
void k_xpool(const float* __restrict__ s, const float* __restrict__ x,
             float* __restrict__ xp) {
    const int lane = threadIdx.x & 31;
    const int w    = threadIdx.x >> 5;            // 0..7  (D tile)
    const int half = lane >> 4, l = lane & 15;
    const int blk  = blockIdx.x;
    const int b    = blk / (V_ / VCHUNK);
    const int vb   = (blk - b * (V_ / VCHUNK)) * VCHUNK;
    const int n0   = w * 16;

    v8f acc0 = {}, acc1 = {}, acc2 = {}, acc3 = {};
    #pragma unroll 2
    for (int vi = 0; vi < VCHUNK; vi += 4) {
        const int va = vb + vi + 2 * half;
        const float* sr0 = s + ((size_t)b * V_ + va) * K_ + l;
        const float* sr1 = sr0 + K_;
        const float* xr0 = x + ((size_t)b * V_ + va) * D_ + n0 + l;
        v2f bm; bm.x = xr0[0];  bm.y = xr0[D_];
        v2f a0; a0.x = sr0[0];  a0.y = sr1[0];
        v2f a1; a1.x = sr0[16]; a1.y = sr1[16];
        v2f a2; a2.x = sr0[32]; a2.y = sr1[32];
        v2f a3; a3.x = sr0[48]; a3.y = sr1[48];
        acc0 = wmma4(a0, bm, acc0);
        acc1 = wmma4(a1, bm, acc1);
        acc2 = wmma4(a2, bm, acc2);
        acc3 = wmma4(a3, bm, acc3);
    }
    #pragma unroll
    for (int r = 0; r < 8; ++r) {
        const int Mb = r + 8 * half;
        float* dst = xp + ((size_t)b * K_ + Mb) * D_ + n0 + l;
        unsafeAtomicAdd(dst + 0 * 16 * D_, acc0[r]);
        unsafeAtomicAdd(dst + 1 * 16 * D_, acc1[r]);
        unsafeAtomicAdd(dst + 2 * 16 * D_, acc2[r]);
        unsafeAtomicAdd(dst + 3 * 16 * D_, acc3[r]);
    }
}

// ---------------------------------------------------------------------------
// K4: out_adj_acc += sum_e w_e * s[row_e] (x) s[col_e]  (gather-GEMM, K=4/step)
// 4 waves/block: wave w owns N tile w. Hot loop (~10M wmma total).
// ---------------------------------------------------------------------------
__global__ __launch_bounds__(128)
void k_edge(const float* __restrict__ s, const int* __restrict__ ei,
            const float* __restrict__ ew, float* __restrict__ adj) {
    const int lane = threadIdx.x & 31;
    const int w    = threadIdx.x >> 5;            // 0..3
    const int half = lane >> 4, l = lane & 15;
    const int blk  = blockIdx.x;
    const int b    = blk / (E_ / ECHUNK);
    const int eb   = (blk - b * (E_ / ECHUNK)) * ECHUNK;
    const int n0   = w * 16;
    const int* rowi = ei;
    const int* coli = ei + E_;
    const float* sb = s + (size_t)b * V_ * K_;

    v8f acc0 = {}, acc1 = {}, acc2 = {}, acc3 = {};
    for (int e = 0; e < ECHUNK; e += 4) {
        const int eL = eb + e + 2 * half;
        const int r0 = rowi[eL], r1 = rowi[eL + 1];
        const int c0 = coli[eL], c1 = coli[eL + 1];
        const float w0 = ew[eL], w1 = ew[eL + 1];
        const float* t0 = sb + (size_t)r0 * K_ + l;
        const float* t1 = sb + (size_t)r1 * K_ + l;
        v2f a0; a0.x = w0 * t0[0];  a0.y = w1 * t1[0];
        v2f a1; a1.x = w0 * t0[16]; a1.y = w1 * t1[16];
        v2f a2; a2.x = w0 * t0[32]; a2.y = w1 * t1[32];
        v2f a3; a3.x = w0 * t0[48]; a3.y = w1 * t1[48];
        v2f bm; bm.x = sb[(size_t)c0 * K_ + n0 + l];
                bm.y = sb[(size_t)c1 * K_ + n0 + l];
        acc0 = wmma4(a0, bm, acc0);
        acc1 = wmma4(a1, bm, acc1);
        acc2 = wmma4(a2, bm, acc2);
        acc3 = wmma4(a3, bm, acc3);
    }
    #pragma unroll
    for (int r = 0; r < 8; ++r) {
        const int Mb = r + 8 * half;
        float* dst = adj + ((size_t)b * K_ + Mb) * K_ + n0 + l;
        unsafeAtomicAdd(dst + 0 * 16 * K_, acc0[r]);
        unsafeAtomicAdd(dst + 1 * 16 * K_, acc1[r]);
        unsafeAtomicAdd(dst + 2 * 16 * K_, acc2[r]);
        unsafeAtomicAdd(dst + 3 * 16 * K_, acc3[r]);
    }
}

// ---------------------------------------------------------------------------
// K5: ss_acc += s^T s over a V-chunk. 4 waves/block (N tile per wave).
// ---------------------------------------------------------------------------
__global__ __launch_bounds__(128)
void k_ssT(const float* __restrict__ s, float* __restrict__ ss) {
    const int lane = threadIdx.x & 31;
    const int w    = threadIdx.x >> 5;            // 0..3
    const int half = lane >> 4, l = lane & 15;
    const int blk  = blockIdx.x;
    const int b    = blk / (V_ / VCHUNK);
    const int vb   = (blk - b * (V_ / VCHUNK)) * VCHUNK;
    const int n0   = w * 16;

    v8f acc0 = {}, acc1 = {}, acc2 = {}, acc3 = {};
    #pragma unroll 2
    for (int vi = 0; vi < VCHUNK; vi += 4) {
        const int va = vb + vi + 2 * half;
        const float* sr0 = s + ((size_t)b * V_ + va) * K_;
        const float* sr1 = sr0 + K_;
        v2f bm; bm.x = sr0[n0 + l]; bm.y = sr1[n0 + l];
        v2f a0; a0.x = sr0[l];      a0.y = sr1[l];
        v2f a1; a1.x = sr0[16 + l]; a1.y = sr1[16 + l];
        v2f a2; a2.x = sr0[32 + l]; a2.y = sr1[32 + l];
        v2f a3; a3.x = sr0[48 + l]; a3.y = sr1[48 + l];
        acc0 = wmma4(a0, bm, acc0);
        acc1 = wmma4(a1, bm, acc1);
        acc2 = wmma4(a2, bm, acc2);
        acc3 = wmma4(a3, bm, acc3);
    }
    #pragma unroll
    for (int r = 0; r < 8; ++r) {
        const int Mb = r + 8 * half;
        float* dst = ss + ((size_t)b * K_ + Mb) * K_ + n0 + l;
        unsafeAtomicAdd(dst + 0 * 16 * K_, acc0[r]);
        unsafeAtomicAdd(dst + 1 * 16 * K_, acc1[r]);
        unsafeAtomicAdd(dst + 2 * 16 * K_, acc2[r]);
        unsafeAtomicAdd(dst + 3 * 16 * K_, acc3[r]);
    }
}

// ---------------------------------------------------------------------------
// K6: per-batch finalize (trace, diag-zero + sqrt-rowsum normalize, ortho
// loss, mincut denominator dot product).  One 256-thread block per batch.
// ---------------------------------------------------------------------------
__device__ __forceinline__ float block_reduce_256(float v, float* red) {
    const int tid = threadIdx.x;
    red[tid] = v;
    __syncthreads();
    #pragma unroll
    for (int off = 128; off > 0; off >>= 1) {
        if (tid < off) red[tid] += red[tid + off];
        __syncthreads();
    }
    float r = red[0];
    __syncthreads();
    return r;
}

__global__ __launch_bounds__(256)
void k_finalize(const float* __restrict__ adj_acc, const float* __restrict__ ss_acc,
                const float* __restrict__ dflat, const float* __restrict__ rss,
                float* __restrict__ adj_out, float* __restrict__ mc_part,
                float* __restrict__ o_part) {
    __shared__ float sm[K_ * K_];
    __shared__ float red[256];
    __shared__ float dd[K_];
    const int tid = threadIdx.x;
    const int b   = blockIdx.x;

    // load pooled adjacency, accumulate trace
    float tr = 0.0f;
    for (int i = tid; i < K_ * K_; i += 256) {
        float v = adj_acc[(size_t)b * K_ * K_ + i];
        sm[i] = v;
        if ((i >> 6) == (i & 63)) tr += v;
    }
    float trace = block_reduce_256(tr, red);

    // zero diagonal, degree-normalize
    if (tid < K_) sm[tid * K_ + tid] = 0.0f;
    __syncthreads();
    if (tid < K_) {
        float rsum = 0.0f;
        #pragma unroll 8
        for (int j = 0; j < K_; ++j) rsum += sm[tid * K_ + j];
        dd[tid] = sqrtf(rsum) + 1e-15f;
    }
    __syncthreads();
    for (int i = tid; i < K_ * K_; i += 256) {
        const int r = i >> 6, c = i & 63;
        adj_out[(size_t)b * K_ * K_ + i] = sm[i] / (dd[r] * dd[c]);
    }

    // orthogonality loss: || ss/||ss||_F - I/sqrt(K) ||_F
    float fs = 0.0f;
    for (int i = tid; i < K_ * K_; i += 256) {
        float v = ss_acc[(size_t)b * K_ * K_ + i];
        fs += v * v;
    }
    const float invf = 1.0f / sqrtf(block_reduce_256(fs, red));
    float ds = 0.0f;
    for (int i = tid; i < K_ * K_; i += 256) {
        float v = ss_acc[(size_t)b * K_ * K_ + i] * invf;
        if ((i >> 6) == (i & 63)) v -= 0.125f;   // 1/sqrt(64)
        ds += v * v;
    }
    const float oloss = sqrtf(block_reduce_256(ds, red));

    // mincut denominator: sum_v d_flat[v] * sum_k s[v,k]^2
    float den = 0.0f;
    for (int v = tid; v < V_; v += 256) den += dflat[v] * rss[(size_t)b * V_ + v];
    den = block_reduce_256(den, red);

    if (tid == 0) {
        mc_part[b] = -(trace / den);
        o_part[b]  = oloss;
    }
}

// K7: fixed-order batch sum of the two scalar losses (deterministic)
__global__ void k_losses(const float* __restrict__ mc_part, const float* __restrict__ o_part,
                         float* __restrict__ out_mc, float* __restrict__ out_o) {
    if (blockIdx.x == 0 && threadIdx.x == 0) {
        float m = 0.0f, o = 0.0f;
        for (int b = 0; b < B_; ++b) { m += mc_part[b]; o += o_part[b]; }
        *out_mc = m; *out_o = o;
    }
}

// ---------------------------------------------------------------------------
extern "C" void kernel_launch(void* const* d_in, const int* in_sizes, int n_in,
                              void* d_out, int out_size, void* d_ws, size_t ws_size,
                              hipStream_t stream) {
    const float* x  = (const float*)d_in[0];
    const int*   ei = (const int*)d_in[1];   // edge_index [2, E]
    const float* ew = (const float*)d_in[2];
    const float* Wm = (const float*)d_in[3];
    const float* bv = (const float*)d_in[4];

    float* out_s   = (float*)d_out;                       // [B,V,K]
    float* out_xp  = out_s  + (size_t)B_ * V_ * K_;       // [B,K,D]
    float* out_adj = out_xp + (size_t)B_ * K_ * D_;       // [B,K,K]
    float* out_mc  = out_adj + (size_t)B_ * K_ * K_;      // scalar
    float* out_o   = out_mc + 1;                          // scalar

    float* ws_dflat = (float*)d_ws;                       // [V]
    float* ws_rss   = ws_dflat + V_;                      // [B,V]
    float* ws_adj   = ws_rss + (size_t)B_ * V_;           // [B,K,K]
    float* ws_ss    = ws_adj + (size_t)B_ * K_ * K_;      // [B,K,K]
    float* ws_mc    = ws_ss + (size_t)B_ * K_ * K_;       // [B]
    float* ws_o     = ws_mc + B_;                         // [B]

    const int initN = (int)((WS_FLOATS + 255) / 256);
    k_init<<<initN, 256, 0, stream>>>((float*)d_ws, out_xp);
    k_dflat<<<(E_ + 255) / 256, 256, 0, stream>>>(ei, ew, ws_dflat);
    k_score_softmax<<<B_ * (V_ / 16), 32, 0, stream>>>(x, Wm, bv, out_s, ws_rss);
    k_xpool<<<B_ * (V_ / VCHUNK), 256, 0, stream>>>(out_s, x, out_xp);
    k_edge<<<B_ * (E_ / ECHUNK), 128, 0, stream>>>(out_s, ei, ew, ws_adj);
    k_ssT<<<B_ * (V_ / VCHUNK), 128, 0, stream>>>(out_s, ws_ss);
    k_finalize<<<B_, 256, 0, stream>>>(ws_adj, ws_ss, ws_dflat, ws_rss, out_adj, ws_mc, ws_o);
    k_losses<<<1, 32, 0, stream>>>(ws_mc, ws_o, out_mc, out_o);
}